// ImprovedDHSA_85083302133881
// MI455X (gfx1250) — compile-verified
//
#include <hip/hip_runtime.h>

// ===== CDNA5 (gfx1250) implementation of ImprovedDHSA =====
// fp32 throughout (matches reference numerics; workload is bandwidth-bound so
// the fp32 WMMA path V_WMMA_F32_16X16X4_F32 is the right precision choice).

typedef __attribute__((ext_vector_type(2))) float v2f;
typedef __attribute__((ext_vector_type(8))) float v8f;

#define DEVINL __device__ __forceinline__

static constexpr int B = 2, C = 64, H = 256, W = 256;
static constexpr int HW = H * W;        // 65536
static constexpr int CHF = C / 2;       // 32 (sorted half)
static constexpr int QKVC = 5 * C;      // 320
static constexpr int HEADS = 4;
static constexpr int D = 64;            // cph * factor = 16*4
static constexpr int NPH = HW / HEADS;  // 16384
static constexpr size_t ARR = (size_t)B * HEADS * D * NPH;  // 8388608 per head-layout array

// D = A(16x4) * B(4x16) + C  (fp32 WMMA). VGPR layouts per CDNA5 ISA 7.12.2:
//  A: lane r<16 -> row m0+r, v0 = K{0|2}, v1 = K{1|3} (K split by lane half)
//  B: lanes stripe N (n0 + (lane&15)), v0 = rows K{0|2}, v1 = rows K{1|3}
//  C/D: vgpr i -> row m0 + i + 8*(lane>=16), col n0 + (lane&15)
DEVINL v8f wmma4(v2f a, v2f b, v8f c) {
  return __builtin_amdgcn_wmma_f32_16x16x4_f32(false, a, false, b, (short)0, c,
                                               false, false);
}

// ---------------------------------------------------------------------------
// 1) argsort 256 along H (first 32 channels): one block per (b,c,w) column
// ---------------------------------------------------------------------------
__global__ void sort_h_kernel(const float* __restrict__ x, float* __restrict__ xw,
                              int* __restrict__ idx_h) {
  __shared__ float sk[256];
  __shared__ int si[256];
  const int wcol = blockIdx.x, c = blockIdx.y, b = blockIdx.z;
  const int t = threadIdx.x;
  const size_t plane = (size_t)(b * C + c) * HW;
  sk[t] = x[plane + (size_t)t * W + wcol];
  si[t] = t;
  __syncthreads();
  for (int k = 2; k <= 256; k <<= 1)
    for (int j = k >> 1; j > 0; j >>= 1) {
      int p = t ^ j;
      if (p > t) {
        bool up = ((t & k) == 0);
        float ka = sk[t], kb = sk[p];
        if ((ka > kb) == up) {
          sk[t] = kb; sk[p] = ka;
          int ti = si[t]; si[t] = si[p]; si[p] = ti;
        }
      }
      __syncthreads();
    }
  xw[plane + (size_t)t * W + wcol] = sk[t];
  idx_h[(size_t)(b * CHF + c) * HW + (size_t)t * W + wcol] = si[t];
}

// ---------------------------------------------------------------------------
// 2) argsort 256 along W (in place on H-sorted values)
// ---------------------------------------------------------------------------
__global__ void sort_w_kernel(float* __restrict__ xw, int* __restrict__ idx_w) {
  __shared__ float sk[256];
  __shared__ int si[256];
  const int h = blockIdx.x, c = blockIdx.y, b = blockIdx.z;
  const int t = threadIdx.x;
  const size_t rowoff = ((size_t)(b * C + c) * H + h) * W;
  sk[t] = xw[rowoff + t];
  si[t] = t;
  __syncthreads();
  for (int k = 2; k <= 256; k <<= 1)
    for (int j = k >> 1; j > 0; j >>= 1) {
      int p = t ^ j;
      if (p > t) {
        bool up = ((t & k) == 0);
        float ka = sk[t], kb = sk[p];
        if ((ka > kb) == up) {
          sk[t] = kb; sk[p] = ka;
          int ti = si[t]; si[t] = si[p]; si[p] = ti;
        }
      }
      __syncthreads();
    }
  xw[rowoff + t] = sk[t];
  idx_w[(size_t)(b * CHF + c) * HW + (size_t)h * W + t] = si[t];
}

// copy channels [32,64) of x into x_work unchanged (float4)
__global__ void copy_upper_kernel(const float* __restrict__ x, float* __restrict__ xw) {
  const size_t per = (size_t)CHF * HW / 4;  // 524288 float4 per batch
  size_t gid = (size_t)blockIdx.x * 256 + threadIdx.x;
  int b = (int)(gid / per);
  size_t rest = gid - (size_t)b * per;
  size_t off = (size_t)(b * C + CHF) * HW / 4 + rest;
  ((float4*)xw)[off] = ((const float4*)x)[off];
}

// ---------------------------------------------------------------------------
// 3) qkv = conv1x1: GEMM  W(320x64) * X(64x65536)  per batch, WMMA f32
//    block = 8 waves -> 4 M-tiles x 2 N-tiles (64 x 32 output / block)
// ---------------------------------------------------------------------------
__global__ void conv1x1_qkv_kernel(const float* __restrict__ xw,
                                   const float* __restrict__ wq,
                                   float* __restrict__ outp) {
  const int b = blockIdx.z;
  const int wave = threadIdx.x >> 5, lane = threadIdx.x & 31;
  const int r = lane & 15, half = lane >> 4;
  const int m0 = (blockIdx.y * 4 + (wave & 3)) * 16;
  const int n0 = (blockIdx.x * 2 + (wave >> 2)) * 16;
  const float* X = xw + (size_t)b * C * HW;
  v8f acc = {};
#pragma unroll
  for (int kk = 0; kk < C; kk += 4) {
    v2f a = *(const v2f*)(wq + (size_t)(m0 + r) * C + kk + 2 * half);
    v2f bb;
    bb.x = X[(size_t)(kk + 2 * half) * HW + n0 + r];
    bb.y = X[(size_t)(kk + 2 * half + 1) * HW + n0 + r];
    acc = wmma4(a, bb, acc);
  }
  float* O = outp + (size_t)b * QKVC * HW;
#pragma unroll
  for (int i = 0; i < 8; ++i)
    O[(size_t)(m0 + i + 8 * half) * HW + n0 + r] = acc[i];
}

// ---------------------------------------------------------------------------
// 4) depthwise 3x3 SAME
// ---------------------------------------------------------------------------
__global__ void dwconv3x3_kernel(const float* __restrict__ in,
                                 const float* __restrict__ w,
                                 float* __restrict__ out) {
  size_t gid = (size_t)blockIdx.x * 256 + threadIdx.x;  // B*QKVC*HW
  int pos = (int)(gid & (HW - 1));
  int chf = (int)(gid >> 16);  // b*QKVC + ch
  int ch = chf % QKVC;
  int i = pos >> 8, j = pos & 255;
  const float* base = in + (gid - pos);
  float acc = 0.f;
#pragma unroll
  for (int di = -1; di <= 1; ++di) {
    int ii = i + di;
    if (ii < 0 || ii >= H) continue;
#pragma unroll
    for (int dj = -1; dj <= 1; ++dj) {
      int jj = j + dj;
      if (jj < 0 || jj >= W) continue;
      acc += base[ii * W + jj] * w[ch * 9 + (di + 1) * 3 + (dj + 1)];
    }
  }
  out[gid] = acc;
}

// ---------------------------------------------------------------------------
// 5) 65536-element bitonic argsort per (b,c) on v.
//    Local phase sorts 8192-element chunks entirely in LDS (64 KB: possible
//    because CDNA5 WGP has 320 KB LDS). Then only 6 global passes + 3 fused
//    LDS merge passes complete n=65536.
// ---------------------------------------------------------------------------
__global__ void vsort_local_kernel(const float* __restrict__ qkv,
                                   float* __restrict__ keys, int* __restrict__ idxb) {
  __shared__ float sk[8192];
  __shared__ int si[8192];
  const int arr = blockIdx.y;            // b*64 + c
  const int cb = blockIdx.x * 8192;
  const int b = arr >> 6, c = arr & 63;
  const float* src = qkv + (size_t)(b * QKVC + 4 * C + c) * HW;
  const int t = threadIdx.x;
  for (int s = t; s < 8192; s += 512) { sk[s] = src[cb + s]; si[s] = cb + s; }
  __syncthreads();
  for (int k = 2; k <= 8192; k <<= 1)
    for (int j = k >> 1; j > 0; j >>= 1) {
      for (int s = t; s < 8192; s += 512) {
        int p = s ^ j;
        if (p > s) {
          bool up = (((cb + s) & k) == 0);
          float ka = sk[s], kb = sk[p];
          if ((ka > kb) == up) {
            sk[s] = kb; sk[p] = ka;
            int ti = si[s]; si[s] = si[p]; si[p] = ti;
          }
        }
      }
      __syncthreads();
    }
  const size_t base = (size_t)arr * HW + cb;
  for (int s = t; s < 8192; s += 512) { keys[base + s] = sk[s]; idxb[base + s] = si[s]; }
}

__global__ void vsort_global_kernel(float* __restrict__ keys, int* __restrict__ idxb,
                                    int k, int j) {
  size_t tid = (size_t)blockIdx.x * 256 + threadIdx.x;  // 128 * 32768
  int arr = (int)(tid >> 15);
  int p = (int)(tid & 32767);
  int i = ((p & ~(j - 1)) << 1) | (p & (j - 1));
  int l = i | j;
  bool up = ((i & k) == 0);
  float* kb = keys + (size_t)arr * HW;
  int* ib = idxb + (size_t)arr * HW;
  float ka = kb[i], kl = kb[l];
  if ((ka > kl) == up) {
    kb[i] = kl; kb[l] = ka;
    int ti = ib[i]; ib[i] = ib[l]; ib[l] = ti;
  }
}

__global__ void vsort_merge_kernel(float* __restrict__ keys, int* __restrict__ idxb,
                                   int k) {
  __shared__ float sk[8192];
  __shared__ int si[8192];
  const int arr = blockIdx.y;
  const int cb = blockIdx.x * 8192;
  const size_t base = (size_t)arr * HW + cb;
  const int t = threadIdx.x;
  for (int s = t; s < 8192; s += 512) { sk[s] = keys[base + s]; si[s] = idxb[base + s]; }
  __syncthreads();
  for (int j = 4096; j > 0; j >>= 1) {
    for (int s = t; s < 8192; s += 512) {
      int p = s ^ j;
      if (p > s) {
        bool up = (((cb + s) & k) == 0);
        float ka = sk[s], kb = sk[p];
        if ((ka > kb) == up) {
          sk[s] = kb; sk[p] = ka;
          int ti = si[s]; si[s] = si[p]; si[p] = ti;
        }
      }
    }
    __syncthreads();
  }
  for (int s = t; s < 8192; s += 512) { keys[base + s] = sk[s]; idxb[base + s] = si[s]; }
}

// ---------------------------------------------------------------------------
// 6) gather q1/k1/q2/k2 through idx + rearrange into head layout (b,h,64,16384)
//    box:    d = ci*4 + (s>>14), n = s & 16383
//    nonbox: d = ci*4 + (s&3),   n = s >> 2
// ---------------------------------------------------------------------------
__global__ void build_heads_kernel(const float* __restrict__ qkv,
                                   const float* __restrict__ skeys,
                                   const int* __restrict__ sidx,
                                   float* __restrict__ qkh,  // [q1h|k1h|q2h|k2h]
                                   float* __restrict__ v1h, float* __restrict__ v2h) {
  size_t gid = (size_t)blockIdx.x * 256 + threadIdx.x;  // B*C*HW
  int s = (int)(gid & 65535);
  int c = (int)((gid >> 16) & 63);
  int b = (int)(gid >> 22);
  int g = sidx[gid];
  const float* base = qkv + (size_t)b * QKVC * HW;
  float q1 = base[(size_t)(0 * C + c) * HW + g];
  float k1 = base[(size_t)(1 * C + c) * HW + g];
  float q2 = base[(size_t)(2 * C + c) * HW + g];
  float k2 = base[(size_t)(3 * C + c) * HW + g];
  float v = skeys[gid];
  int head = c >> 4, ci = c & 15;
  int f = s >> 14, n = s & 16383;
  size_t rowb = ((size_t)(b * HEADS + head) * D + ci * 4 + f) * NPH + n;
  int f2 = s & 3, n2 = s >> 2;
  size_t rown = ((size_t)(b * HEADS + head) * D + ci * 4 + f2) * NPH + n2;
  qkh[rowb] = q1;
  qkh[ARR + rowb] = k1;
  qkh[2 * ARR + rown] = q2;
  qkh[3 * ARR + rown] = k2;
  v1h[rowb] = v;
  v2h[rown] = v;
}

// ---------------------------------------------------------------------------
// 7) L2-normalize each 16384-element row of q1h/k1h/q2h/k2h (contiguous)
// ---------------------------------------------------------------------------
__global__ void l2norm_rows_kernel(float* __restrict__ qk) {
  __shared__ float red[8];
  __shared__ float scale_s;
  float* p = qk + (size_t)blockIdx.x * NPH;
  const int t = threadIdx.x;
  float s = 0.f;
  for (int i = t; i < NPH; i += 256) { float v = p[i]; s += v * v; }
  for (int off = 16; off; off >>= 1) s += __shfl_xor(s, off);
  if ((t & 31) == 0) red[t >> 5] = s;
  __syncthreads();
  if (t == 0) {
    float tot = 0.f;
    for (int i = 0; i < 8; ++i) tot += red[i];
    scale_s = 1.0f / fmaxf(sqrtf(tot), 1e-12f);
  }
  __syncthreads();
  float sc = scale_s;
  for (int i = t; i < NPH; i += 256) p[i] *= sc;
}

// ---------------------------------------------------------------------------
// 8) attn partials: per (combo, chunk) compute Q(64xK) * K(64xK)^T over
//    K-chunk of 1024, staging 64x64 sub-tiles of Q and K in LDS.
//    combo = b*8 + var*4 + head ; 16 chunks of 1024.
// ---------------------------------------------------------------------------
__global__ void attn_qk_kernel(const float* __restrict__ qk, float* __restrict__ attnP) {
  __shared__ float lq[64 * 64];
  __shared__ float lk[64 * 64];
  const int combo = blockIdx.y;
  const int b = combo >> 3, var = (combo >> 2) & 1, head = combo & 3;
  const size_t rb = (size_t)(b * HEADS + head) * D * NPH;
  const float* Q = qk + (size_t)(var * 2) * ARR + rb;
  const float* Kp = qk + (size_t)(var * 2 + 1) * ARR + rb;
  const int kbase = blockIdx.x * 1024;
  const int t = threadIdx.x;
  const int wave = t >> 5, lane = t & 31, r = lane & 15, half = lane >> 4;
  const int m0 = (wave >> 2) * 16, e0 = (wave & 3) * 16;
  v8f acc = {};
  for (int kb = kbase; kb < kbase + 1024; kb += 64) {
    for (int s = t; s < 1024; s += 512) {  // 1024 float4 slots per tile
      int row = s >> 4, c4 = (s & 15) << 2;
      *(float4*)&lq[row * 64 + c4] = *(const float4*)&Q[(size_t)row * NPH + kb + c4];
      *(float4*)&lk[row * 64 + c4] = *(const float4*)&Kp[(size_t)row * NPH + kb + c4];
      if (kb + 64 < kbase + 1024) {  // gfx1250 global_prefetch of next tile
        __builtin_prefetch(&Q[(size_t)row * NPH + kb + 64 + c4], 0, 0);
        __builtin_prefetch(&Kp[(size_t)row * NPH + kb + 64 + c4], 0, 0);
      }
    }
    __syncthreads();
#pragma unroll
    for (int kk = 0; kk < 64; kk += 4) {
      v2f a = *(const v2f*)&lq[(m0 + r) * 64 + kk + 2 * half];
      v2f bb = *(const v2f*)&lk[(e0 + r) * 64 + kk + 2 * half];
      acc = wmma4(a, bb, acc);
    }
    __syncthreads();
  }
  float* P = attnP + ((size_t)combo * 16 + blockIdx.x) * 4096;
#pragma unroll
  for (int i = 0; i < 8; ++i)
    P[(m0 + i + 8 * half) * 64 + e0 + r] = acc[i];
}

// ---------------------------------------------------------------------------
// 9) reduce chunks, *temperature, softmax_1 (denominator + 1)
// ---------------------------------------------------------------------------
__global__ void attn_softmax_kernel(const float* __restrict__ attnP,
                                    const float* __restrict__ temperature,
                                    float* __restrict__ attnF) {
  __shared__ float att[4096];
  __shared__ float rs[64];
  const int combo = blockIdx.x;
  const int head = combo & 3;
  const int t = threadIdx.x;
  const float temp = temperature[head];
  const float* P = attnP + (size_t)combo * 16 * 4096;
  for (int e = t; e < 4096; e += 256) {
    float s = 0.f;
    for (int chk = 0; chk < 16; ++chk) s += P[chk * 4096 + e];
    att[e] = expf(s * temp);
  }
  __syncthreads();
  if (t < 64) {
    float s = 0.f;
    for (int e = 0; e < 64; ++e) s += att[t * 64 + e];
    rs[t] = 1.0f / (s + 1.0f);
  }
  __syncthreads();
  float* F = attnF + (size_t)combo * 4096;
  for (int e = t; e < 4096; e += 256) F[e] = att[e] * rs[e >> 6];
}

// ---------------------------------------------------------------------------
// 10) out = attn(64x64) @ V(64x16384); attn cached in LDS, WMMA K-loop
// ---------------------------------------------------------------------------
__global__ void attn_out_kernel(const float* __restrict__ attnF,
                                const float* __restrict__ v1h,
                                const float* __restrict__ v2h,
                                float* __restrict__ out1h, float* __restrict__ out2h) {
  __shared__ float la[4096];
  const int combo = blockIdx.y;
  const int b = combo >> 3, var = (combo >> 2) & 1, head = combo & 3;
  const size_t rb = (size_t)(b * HEADS + head) * D * NPH;
  const float* V = (var ? v2h : v1h) + rb;
  float* O = (var ? out2h : out1h) + rb;
  const float* A = attnF + (size_t)combo * 4096;
  const int t = threadIdx.x;
  for (int s = t; s < 1024; s += 256) *(float4*)&la[s * 4] = *(const float4*)&A[s * 4];
  __syncthreads();
  const int wave = t >> 5, lane = t & 31, r = lane & 15, half = lane >> 4;
  const int m0 = (wave & 3) * 16;
  const int n0 = blockIdx.x * 32 + (wave >> 2) * 16;
  v8f acc = {};
#pragma unroll
  for (int kk = 0; kk < 64; kk += 4) {
    v2f a = *(const v2f*)&la[(m0 + r) * 64 + kk + 2 * half];
    v2f bb;
    bb.x = V[(size_t)(kk + 2 * half) * NPH + n0 + r];
    bb.y = V[(size_t)(kk + 2 * half + 1) * NPH + n0 + r];
    acc = wmma4(a, bb, acc);
  }
#pragma unroll
  for (int i = 0; i < 8; ++i)
    O[(size_t)(m0 + i + 8 * half) * NPH + n0 + r] = acc[i];
}

// ---------------------------------------------------------------------------
// 11) out1*out2 in sorted domain, scatter through idx back to original order
// ---------------------------------------------------------------------------
__global__ void combine_scatter_kernel(const float* __restrict__ out1h,
                                       const float* __restrict__ out2h,
                                       const int* __restrict__ sidx,
                                       float* __restrict__ prodbuf) {
  size_t gid = (size_t)blockIdx.x * 256 + threadIdx.x;  // B*C*HW
  int s = (int)(gid & 65535);
  int c = (int)((gid >> 16) & 63);
  int b = (int)(gid >> 22);
  int head = c >> 4, ci = c & 15;
  int f = s >> 14, n = s & 16383;
  float o1 = out1h[((size_t)(b * HEADS + head) * D + ci * 4 + f) * NPH + n];
  int f2 = s & 3, n2 = s >> 2;
  float o2 = out2h[((size_t)(b * HEADS + head) * D + ci * 4 + f2) * NPH + n2];
  int g = sidx[gid];
  prodbuf[(gid - s) + g] = o1 * o2;
}

// ---------------------------------------------------------------------------
// 12) proj conv1x1: GEMM 64x64 * 64x65536 per batch, WMMA f32
// ---------------------------------------------------------------------------
__global__ void conv1x1_proj_kernel(const float* __restrict__ prod,
                                    const float* __restrict__ pw,
                                    float* __restrict__ outp) {
  const int b = blockIdx.z;
  const int wave = threadIdx.x >> 5, lane = threadIdx.x & 31;
  const int r = lane & 15, half = lane >> 4;
  const int m0 = (wave & 3) * 16;
  const int n0 = (blockIdx.x * 2 + (wave >> 2)) * 16;
  const float* X = prod + (size_t)b * C * HW;
  v8f acc = {};
#pragma unroll
  for (int kk = 0; kk < C; kk += 4) {
    v2f a = *(const v2f*)(pw + (size_t)(m0 + r) * C + kk + 2 * half);
    v2f bb;
    bb.x = X[(size_t)(kk + 2 * half) * HW + n0 + r];
    bb.y = X[(size_t)(kk + 2 * half + 1) * HW + n0 + r];
    acc = wmma4(a, bb, acc);
  }
  float* O = outp + (size_t)b * C * HW;
#pragma unroll
  for (int i = 0; i < 8; ++i)
    O[(size_t)(m0 + i + 8 * half) * HW + n0 + r] = acc[i];
}

// ---------------------------------------------------------------------------
// 13) invert W-sort then H-sort for the first 32 channels (scatter form)
// ---------------------------------------------------------------------------
__global__ void final_unsort_kernel(const float* __restrict__ proj,
                                    const int* __restrict__ idx_h,
                                    const int* __restrict__ idx_w,
                                    float* __restrict__ out) {
  size_t gid = (size_t)blockIdx.x * 256 + threadIdx.x;  // B*C*HW
  int pos = (int)(gid & 65535);
  int c = (int)((gid >> 16) & 63);
  int b = (int)(gid >> 22);
  float val = proj[gid];
  if (c < CHF) {
    int i = pos >> 8, j = pos & 255;
    size_t pl = (size_t)(b * CHF + c) * HW;
    int wp = idx_w[pl + (size_t)i * W + j];
    int hp = idx_h[pl + (size_t)i * W + wp];
    out[(size_t)(b * C + c) * HW + (size_t)hp * W + wp] = val;
  } else {
    out[gid] = val;
  }
}

// ===========================================================================
extern "C" void kernel_launch(void* const* d_in, const int* in_sizes, int n_in,
                              void* d_out, int out_size, void* d_ws, size_t ws_size,
                              hipStream_t stream) {
  (void)in_sizes; (void)n_in; (void)out_size; (void)ws_size;
  const float* x = (const float*)d_in[0];
  const float* qkv_w = (const float*)d_in[1];
  const float* dw_w = (const float*)d_in[2];
  const float* proj_w = (const float*)d_in[3];
  const float* temperature = (const float*)d_in[4];
  float* out = (float*)d_out;

  char* wp = (char*)d_ws;
  auto alloc = [&](size_t bytes) {
    char* p = wp;
    wp += (bytes + 255) & ~(size_t)255;
    return p;
  };
  float* x_work = (float*)alloc(sizeof(float) * B * C * HW);
  int* idx_h = (int*)alloc(sizeof(int) * B * CHF * HW);
  int* idx_w = (int*)alloc(sizeof(int) * B * CHF * HW);
  float* qkv_pre = (float*)alloc(sizeof(float) * B * QKVC * HW);
  float* qkvb = (float*)alloc(sizeof(float) * B * QKVC * HW);
  float* skeys = (float*)alloc(sizeof(float) * B * C * HW);
  int* sidx = (int*)alloc(sizeof(int) * B * C * HW);
  float* qkh = (float*)alloc(sizeof(float) * 4 * ARR);  // q1h,k1h,q2h,k2h
  float* v1h = (float*)alloc(sizeof(float) * ARR);
  float* v2h = (float*)alloc(sizeof(float) * ARR);
  float* attnP = (float*)alloc(sizeof(float) * 16 * 16 * 4096);
  float* attnF = (float*)alloc(sizeof(float) * 16 * 4096);
  float* out1h = (float*)alloc(sizeof(float) * ARR);
  float* out2h = (float*)alloc(sizeof(float) * ARR);
  float* prodbuf = (float*)alloc(sizeof(float) * B * C * HW);
  float* projbuf = (float*)alloc(sizeof(float) * B * C * HW);

  // 1-2) spatial sorts of first 32 channels, 3) copy rest
  sort_h_kernel<<<dim3(W, CHF, B), 256, 0, stream>>>(x, x_work, idx_h);
  sort_w_kernel<<<dim3(H, CHF, B), 256, 0, stream>>>(x_work, idx_w);
  copy_upper_kernel<<<(B * CHF * HW / 4) / 256, 256, 0, stream>>>(x, x_work);

  // 4) qkv conv (WMMA) + depthwise
  conv1x1_qkv_kernel<<<dim3(HW / 32, QKVC / 64, B), 256, 0, stream>>>(x_work, qkv_w,
                                                                      qkv_pre);
  dwconv3x3_kernel<<<(size_t)B * QKVC * HW / 256, 256, 0, stream>>>(qkv_pre, dw_w,
                                                                    qkvb);

  // 5) 65536-wide argsort of v per (b,c): LDS-fused bitonic
  vsort_local_kernel<<<dim3(HW / 8192, B * C), 512, 0, stream>>>(qkvb, skeys, sidx);
  for (int k = 16384; k <= 65536; k <<= 1) {
    for (int j = k >> 1; j >= 8192; j >>= 1)
      vsort_global_kernel<<<(B * C * HW / 2) / 256, 256, 0, stream>>>(skeys, sidx, k, j);
    vsort_merge_kernel<<<dim3(HW / 8192, B * C), 512, 0, stream>>>(skeys, sidx, k);
  }

  // 6-7) gather into head layout, L2-normalize q/k rows
  build_heads_kernel<<<(size_t)B * C * HW / 256, 256, 0, stream>>>(qkvb, skeys, sidx,
                                                                   qkh, v1h, v2h);
  l2norm_rows_kernel<<<4 * B * HEADS * D, 256, 0, stream>>>(qkh);

  // 8-10) attention (WMMA)
  attn_qk_kernel<<<dim3(16, 16), 512, 0, stream>>>(qkh, attnP);
  attn_softmax_kernel<<<16, 256, 0, stream>>>(attnP, temperature, attnF);
  attn_out_kernel<<<dim3(NPH / 32, 16), 256, 0, stream>>>(attnF, v1h, v2h, out1h,
                                                          out2h);

  // 11-13) combine, proj conv (WMMA), inverse permutations
  combine_scatter_kernel<<<(size_t)B * C * HW / 256, 256, 0, stream>>>(out1h, out2h,
                                                                       sidx, prodbuf);
  conv1x1_proj_kernel<<<dim3(HW / 32, 1, B), 256, 0, stream>>>(prodbuf, proj_w,
                                                               projbuf);
  final_unsort_kernel<<<(size_t)B * C * HW / 256, 256, 0, stream>>>(projbuf, idx_h,
                                                                    idx_w, out);
}